// FactorizedAdaptiveMixing_88630945120260
// MI455X (gfx1250) — compile-verified
//
#include <hip/hip_runtime.h>

typedef __attribute__((ext_vector_type(2))) float v2f;
typedef __attribute__((ext_vector_type(8))) float v8f;
typedef __attribute__((ext_vector_type(4))) unsigned int u32x4;
typedef __attribute__((ext_vector_type(4))) int i32x4;
typedef __attribute__((ext_vector_type(8))) int i32x8;

#define G 4
#define KS 8
#define EIN 64
#define EOUT 64
#define PIN 32
#define POUT 128
#define QD 256
#define BN 2400
#define NT 150            // 16-token tiles
#define FLAT 32768        // G*POUT*EOUT
#define GOUT 8192         // per-group common outputs (EIN*EOUT + PIN*POUT)
#define EPS 1e-5f
#define KC_N 8            // K-split for final projection

// ---- workspace layout (float offsets) ----
#define OFF_W   ((size_t)0)
#define SZ_W    ((size_t)QD * FLAT)            // 8,388,608
#define OFF_M   (OFF_W + SZ_W)
#define SZ_MS   ((size_t)BN * G * 4096)        // 39,321,600
#define OFF_S   (OFF_M + SZ_MS)
#define OFF_O2  (OFF_S + SZ_MS)
#define SZ_O2   ((size_t)BN * FLAT)            // 78,643,200
#define OFF_YP  (OFF_O2 + SZ_O2)               // KC_N * BN * QD

#if defined(__AMDGCN__) && __has_builtin(__builtin_amdgcn_tensor_load_to_lds)
#define HAVE_TDM 1
#endif

__device__ inline v8f wmma4(v2f a, v2f b, v8f c) {
  // D(16x16,f32) = A(16x4,f32) * B(4x16,f32) + C   -> v_wmma_f32_16x16x4_f32
  return __builtin_amdgcn_wmma_f32_16x16x4_f32(false, a, false, b, (short)0, c,
                                               false, false);
}

__device__ inline void tdm_wait0() {
#if __has_builtin(__builtin_amdgcn_s_wait_tensorcnt)
  __builtin_amdgcn_s_wait_tensorcnt(0);
#else
  asm volatile("s_wait_tensorcnt 0x0" ::: "memory");
#endif
}

#ifdef HAVE_TDM
// Issue a TDM 2-D tile load: tile_rows x tile_cols fp32 elements, row stride
// (in elements) row_stride_elts, from global gsrc into LDS at lds_dst.
// D# packing per CDNA5 ISA ch.8 (group0: count/lds/global/type=2; group1:
// data_size=4B, tensor_dim0=tile_dim0=cols, tensor_dim1=tile_dim1=rows,
// tensor_dim0_stride=row_stride).  Groups 2/3 unused (2-D tensor).
__device__ inline void tdm_load_2d(const void* gsrc, void* lds_dst,
                                   unsigned tile_cols, unsigned tile_rows,
                                   unsigned row_stride_elts) {
  const unsigned long long ga = (unsigned long long)(size_t)gsrc;
  u32x4 g0 = {0u, 0u, 0u, 0u};
  g0[0] = 1u;                                        // count=1, user descriptor
  g0[1] = (unsigned)(size_t)lds_dst;                 // LDS byte address
  g0[2] = (unsigned)(ga & 0xffffffffull);            // global_addr[31:0]
  g0[3] = ((unsigned)(ga >> 32) & 0x1ffffffu) | (2u << 30);  // addr hi | type=2
  i32x8 g1 = {0, 0, 0, 0, 0, 0, 0, 0};
  g1[0] = 0x00020000;                                // data_size = 4 bytes
  g1[1] = (int)(tile_cols << 16);                    // tensor_dim0[15:0]
  g1[2] = (int)((tile_cols >> 16) | (tile_rows << 16));  // dim0 hi | tensor_dim1 lo
  g1[3] = (int)((tile_rows >> 16) | (tile_cols << 16));  // dim1 hi | tile_dim0
  g1[4] = (int)tile_rows;                            // tile_dim1 (tile_dim2=0)
  g1[5] = (int)row_stride_elts;                      // tensor_dim0_stride[31:0]
  i32x4 z4 = {0, 0, 0, 0};
#if __clang_major__ >= 23
  i32x8 z8 = {0, 0, 0, 0, 0, 0, 0, 0};
  __builtin_amdgcn_tensor_load_to_lds(g0, g1, z4, z4, z8, 0);
#else
  __builtin_amdgcn_tensor_load_to_lds(g0, g1, z4, z4, 0);
#endif
}
#endif  // HAVE_TDM

__device__ inline void block_reduce_pair(float& s, float& s2, float* red) {
  const int tid = threadIdx.x;
  red[tid] = s;
  red[256 + tid] = s2;
  __syncthreads();
  for (int st = 128; st > 0; st >>= 1) {
    if (tid < st) {
      red[tid] += red[tid + st];
      red[256 + tid] += red[256 + tid + st];
    }
    __syncthreads();
  }
  s = red[0];
  s2 = red[256];
  __syncthreads();
}

// ------------------------------------------------------------------
// Kernel 1: W = base_w + lora_B @ lora_A   (256 x 32768, K = 64)
// One 16x16 output tile per wave.
// ------------------------------------------------------------------
__global__ __launch_bounds__(256) void k_build_w(const float* __restrict__ base_w,
                                                 const float* __restrict__ lora_A,
                                                 const float* __restrict__ lora_B,
                                                 float* __restrict__ W) {
  const int tile = blockIdx.x * 8 + (threadIdx.x >> 5);  // 0..32767
  const int lane = threadIdx.x & 31;
  const int half = lane >> 4;
  const int l = lane & 15;
  const int fT = tile & 2047;  // 2048 column tiles
  const int oT = tile >> 11;   // 16 row tiles
  const int f0 = fT * 16, o0 = oT * 16;
  v8f c = {};
  for (int k = 0; k < 64; k += 4) {
    const int ka = k + half * 2;
    v2f a, b;
    a[0] = lora_B[(o0 + l) * 64 + ka];
    a[1] = lora_B[(o0 + l) * 64 + ka + 1];
    b[0] = lora_A[(size_t)ka * FLAT + f0 + l];
    b[1] = lora_A[(size_t)(ka + 1) * FLAT + f0 + l];
    c = wmma4(a, b, c);
  }
  for (int r = 0; r < 8; ++r) {
    const size_t idx = (size_t)(o0 + r + half * 8) * FLAT + f0 + l;
    W[idx] = c[r] + base_w[idx];
  }
}

// ------------------------------------------------------------------
// Kernel 2: per (16-token tile, group) -> M_all, S_all
//   Mc/Sc: (16 x 8192, K=256) WMMA GEMM vs cpg_w (L2-resident)
//   + bias + low-rank dynamic part ml@(mz)@mr / sl@(sz)@sr
// ------------------------------------------------------------------
__global__ __launch_bounds__(256) void k_gen_ms(const float* __restrict__ query,
                                                const float* __restrict__ cpg_w,
                                                const float* __restrict__ cpg_b,
                                                const float* __restrict__ ppg_w,
                                                const float* __restrict__ ppg_b,
                                                const float* __restrict__ ml,
                                                const float* __restrict__ mr,
                                                const float* __restrict__ sl,
                                                const float* __restrict__ sr,
                                                float* __restrict__ M_all,
                                                float* __restrict__ S_all) {
  __shared__ float pv[16 * 128];     // mz||sz per token
  __shared__ float tmpM[16 * 512];   // (mz @ mr) per token: 8 x 64
  __shared__ float tmpS[16 * 256];   // (sz @ sr) per token: 8 x 32
  const int tid = threadIdx.x;
  const int t0 = blockIdx.x * 16;
  const int g = blockIdx.y;

  // priv = q @ ppg_w.T + ppg_b  (only this group's 128 rows per token)
  for (int i = 0; i < 8; ++i) {
    const int idx = tid + i * 256;           // 0..2047
    const int t = idx >> 7, r = idx & 127;
    const int s = r >> 6, rr = r & 63;
    const int row = (s * G + g) * 64 + rr;
    float acc = ppg_b[row];
    const float* qr = query + (size_t)(t0 + t) * QD;
    const float* wr = ppg_w + (size_t)row * QD;
    for (int k = 0; k < QD; ++k) acc += qr[k] * wr[k];
    pv[idx] = acc;
  }
  __syncthreads();
  // tmpM[t][k][j] = sum_l mz[k][l] * mr[g][l][j]
  for (int i = 0; i < 32; ++i) {
    const int idx = tid + i * 256;  // 0..8191
    const int t = idx >> 9, rem = idx & 511, k = rem >> 6, j = rem & 63;
    float acc = 0.f;
    for (int l2 = 0; l2 < KS; ++l2)
      acc += pv[t * 128 + k * 8 + l2] * mr[g * 512 + l2 * 64 + j];
    tmpM[idx] = acc;
  }
  // tmpS[t][k][p] = sum_l sz[k][l] * sr[g][l][p]
  for (int i = 0; i < 16; ++i) {
    const int idx = tid + i * 256;  // 0..4095
    const int t = idx >> 8, rem = idx & 255, k = rem >> 5, p = rem & 31;
    float acc = 0.f;
    for (int l2 = 0; l2 < KS; ++l2)
      acc += pv[t * 128 + 64 + k * 8 + l2] * sr[g * 256 + l2 * 32 + p];
    tmpS[idx] = acc;
  }
  __syncthreads();

  const int w = tid >> 5, lane = tid & 31, half = lane >> 4, l = lane & 15;
  const float* qrow = query + (size_t)(t0 + l) * QD;  // A row = token l
  for (int nt = 0; nt < 64; ++nt) {
    const int col = w * 1024 + nt * 16 + l;           // 0..8191 within group
    const size_t wrow = (size_t)(g * GOUT + col) * QD;
    v8f c = {};
    for (int k = 0; k < QD; k += 4) {
      const int ka = k + half * 2;
      v2f a, b;
      a[0] = qrow[ka];
      a[1] = qrow[ka + 1];
      b[0] = cpg_w[wrow + ka];
      b[1] = cpg_w[wrow + ka + 1];
      c = wmma4(a, b, c);
    }
    const float bias = cpg_b[g * GOUT + col];
    if (col < 4096) {  // M part: col = i*64 + j
      const int i2 = col >> 6, j2 = col & 63;
      for (int r = 0; r < 8; ++r) {
        const int t = r + half * 8;
        float v = c[r] + bias;
        for (int k2 = 0; k2 < KS; ++k2)
          v += ml[g * 512 + i2 * 8 + k2] * tmpM[t * 512 + k2 * 64 + j2];
        M_all[(size_t)((t0 + t) * G + g) * 4096 + col] = v;
      }
    } else {           // S part: col-4096 = o*32 + p
      const int cc = col - 4096;
      const int o2 = cc >> 5, p2 = cc & 31;
      for (int r = 0; r < 8; ++r) {
        const int t = r + half * 8;
        float v = c[r] + bias;
        for (int k2 = 0; k2 < KS; ++k2)
          v += sl[g * 1024 + o2 * 8 + k2] * tmpS[t * 256 + k2 * 32 + p2];
        S_all[(size_t)((t0 + t) * G + g) * 4096 + cc] = v;
      }
    }
  }
}

// ------------------------------------------------------------------
// Kernel 3: per (token, group): relu(LN(S @ relu(LN(x @ M)))) -> out2
// ------------------------------------------------------------------
__global__ __launch_bounds__(256) void k_mix(const float* __restrict__ x,
                                             const float* __restrict__ M_all,
                                             const float* __restrict__ S_all,
                                             float* __restrict__ out2) {
  __shared__ float xg[PIN * EIN];   // 2048
  __shared__ float ms[4096];        // holds M, later reused for S
  __shared__ float a1[PIN * EOUT];  // 2048
  __shared__ float red[512];
  const int tid = threadIdx.x;
  const int t = blockIdx.x, g = blockIdx.y;
  const size_t baseMS = (size_t)(t * G + g) * 4096;
  const size_t baseX = (size_t)(t * G + g) * (PIN * EIN);
  for (int i = tid; i < PIN * EIN; i += 256) xg[i] = x[baseX + i];
  for (int i = tid; i < 4096; i += 256) ms[i] = M_all[baseMS + i];
  __syncthreads();

  const int w = tid >> 5, lane = tid & 31, half = lane >> 4, l = lane & 15;
  // A1 = x @ M : (32 x 64, K=64); 8 tiles, one per wave
  {
    const int mt = w >> 2, ntd = w & 3;
    v8f c = {};
    for (int k = 0; k < EIN; k += 4) {
      const int ka = k + half * 2;
      v2f a, b;
      a[0] = xg[(mt * 16 + l) * EIN + ka];
      a[1] = xg[(mt * 16 + l) * EIN + ka + 1];
      b[0] = ms[ka * EOUT + ntd * 16 + l];
      b[1] = ms[(ka + 1) * EOUT + ntd * 16 + l];
      c = wmma4(a, b, c);
    }
    for (int r = 0; r < 8; ++r)
      a1[(mt * 16 + r + half * 8) * EOUT + ntd * 16 + l] = c[r];
  }
  __syncthreads();
  // LN + ReLU over 2048
  float s = 0.f, s2 = 0.f;
  for (int i = tid; i < 2048; i += 256) { float v = a1[i]; s += v; s2 += v * v; }
  block_reduce_pair(s, s2, red);
  {
    const float mean = s * (1.f / 2048.f);
    const float var = s2 * (1.f / 2048.f) - mean * mean;
    const float rs = rsqrtf(var + EPS);
    for (int i = tid; i < 2048; i += 256) {
      float v = (a1[i] - mean) * rs;
      a1[i] = v > 0.f ? v : 0.f;
    }
  }
  __syncthreads();
  for (int i = tid; i < 4096; i += 256) ms[i] = S_all[baseMS + i];
  __syncthreads();
  // A2 = S @ A1 : (128 x 64, K=32); wave w = o-tile, 4 d-tiles in registers
  v8f c2[4] = {};
  for (int k = 0; k < PIN; k += 4) {
    const int ka = k + half * 2;
    v2f a;
    a[0] = ms[(w * 16 + l) * PIN + ka];
    a[1] = ms[(w * 16 + l) * PIN + ka + 1];
    for (int dt = 0; dt < 4; ++dt) {
      v2f b;
      b[0] = a1[ka * EOUT + dt * 16 + l];
      b[1] = a1[(ka + 1) * EOUT + dt * 16 + l];
      c2[dt] = wmma4(a, b, c2[dt]);
    }
  }
  // LN + ReLU over 8192 (register-resident)
  s = 0.f; s2 = 0.f;
  for (int dt = 0; dt < 4; ++dt)
    for (int r = 0; r < 8; ++r) { const float v = c2[dt][r]; s += v; s2 += v * v; }
  block_reduce_pair(s, s2, red);
  const float mean = s * (1.f / 8192.f);
  const float var = s2 * (1.f / 8192.f) - mean * mean;
  const float rs = rsqrtf(var + EPS);
  float* outp = out2 + (size_t)t * FLAT + (size_t)g * GOUT;
  for (int dt = 0; dt < 4; ++dt)
    for (int r = 0; r < 8; ++r) {
      float v = (c2[dt][r] - mean) * rs;
      v = v > 0.f ? v : 0.f;
      outp[(w * 16 + r + half * 8) * EOUT + dt * 16 + l] = v;
    }
}

// ------------------------------------------------------------------
// Kernel 4: y_part[kc] = out2[:, kc-chunk] @ W[:, kc-chunk].T
//   (16 tokens x 256 out, K split 8 ways)
//   A tiles (16 x 256 fp32, row stride 32768) DMA'd into double-buffered
//   LDS via the Tensor Data Mover, overlapped with WMMA compute.
// ------------------------------------------------------------------
__global__ __launch_bounds__(256) void k_proj(const float* __restrict__ out2,
                                              const float* __restrict__ W,
                                              float* __restrict__ y_part) {
  __shared__ float as[2][16 * 256];
  const int tid = threadIdx.x;
  const int t0 = blockIdx.x * 16;
  const int kc = blockIdx.y;
  const int w = tid >> 5, lane = tid & 31, half = lane >> 4, l = lane & 15;
  v8f acc[2] = {};
#ifdef HAVE_TDM
  if (w == 0) {
    tdm_load_2d(out2 + (size_t)t0 * FLAT + (size_t)kc * 4096, as[0],
                /*cols=*/256, /*rows=*/16, /*stride=*/FLAT);
    tdm_wait0();
  }
  __syncthreads();
#endif
  for (int ksl = 0; ksl < 16; ++ksl) {
    const size_t kbase = (size_t)kc * 4096 + (size_t)ksl * 256;
    const int buf = ksl & 1;
#ifdef HAVE_TDM
    // Prefetch next A slice into the other LDS buffer while computing.
    if (w == 0 && ksl + 1 < 16) {
      tdm_load_2d(out2 + (size_t)t0 * FLAT + kbase + 256, as[buf ^ 1],
                  /*cols=*/256, /*rows=*/16, /*stride=*/FLAT);
    }
#else
    for (int i = tid; i < 4096; i += 256) {
      const int tt = i >> 8, kk = i & 255;
      as[buf][tt * 256 + kk] = out2[(size_t)(t0 + tt) * FLAT + kbase + kk];
    }
    __syncthreads();
#endif
    for (int k = 0; k < 256; k += 4) {
      const int ka = k + half * 2;
      v2f a;
      a[0] = as[buf][l * 256 + ka];
      a[1] = as[buf][l * 256 + ka + 1];
      const size_t kg = kbase + ka;
      for (int nt = 0; nt < 2; ++nt) {
        const int o = (w * 2 + nt) * 16 + l;
        v2f b;
        b[0] = W[(size_t)o * FLAT + kg];
        b[1] = W[(size_t)o * FLAT + kg + 1];
        acc[nt] = wmma4(a, b, acc[nt]);
      }
    }
#ifdef HAVE_TDM
    if (w == 0 && ksl + 1 < 16) tdm_wait0();
#endif
    __syncthreads();
  }
  for (int nt = 0; nt < 2; ++nt)
    for (int r = 0; r < 8; ++r) {
      const int t = t0 + r + half * 8;
      const int o = (w * 2 + nt) * 16 + l;
      y_part[(size_t)kc * ((size_t)BN * QD) + (size_t)t * QD + o] = acc[nt][r];
    }
}

// ------------------------------------------------------------------
// Kernel 5: sum partials + bias + residual, final LN * gamma + beta
// ------------------------------------------------------------------
__global__ __launch_bounds__(256) void k_final(const float* __restrict__ query,
                                               const float* __restrict__ out_bias,
                                               const float* __restrict__ gamma,
                                               const float* __restrict__ beta,
                                               const float* __restrict__ y_part,
                                               float* __restrict__ out) {
  __shared__ float red[512];
  const int t = blockIdx.x;
  const int e = threadIdx.x;
  float v = query[(size_t)t * QD + e] + out_bias[e];
  for (int kc = 0; kc < KC_N; ++kc)
    v += y_part[(size_t)kc * ((size_t)BN * QD) + (size_t)t * QD + e];
  float s = v, s2 = v * v;
  block_reduce_pair(s, s2, red);
  const float mean = s * (1.f / 256.f);
  const float var = s2 * (1.f / 256.f) - mean * mean;
  out[(size_t)t * QD + e] = (v - mean) * rsqrtf(var + EPS) * gamma[e] + beta[e];
}

// ------------------------------------------------------------------
extern "C" void kernel_launch(void* const* d_in, const int* in_sizes, int n_in,
                              void* d_out, int out_size, void* d_ws, size_t ws_size,
                              hipStream_t stream) {
  const float* x      = (const float*)d_in[1];
  const float* query  = (const float*)d_in[2];
  const float* cpg_w  = (const float*)d_in[3];
  const float* cpg_b  = (const float*)d_in[4];
  const float* ppg_w  = (const float*)d_in[5];
  const float* ppg_b  = (const float*)d_in[6];
  const float* ml     = (const float*)d_in[7];
  const float* mr     = (const float*)d_in[8];
  const float* sl     = (const float*)d_in[9];
  const float* sr     = (const float*)d_in[10];
  const float* base_w = (const float*)d_in[11];
  const float* lora_A = (const float*)d_in[12];
  const float* lora_B = (const float*)d_in[13];
  const float* obias  = (const float*)d_in[14];
  const float* gamma  = (const float*)d_in[15];
  const float* beta   = (const float*)d_in[16];

  float* ws = (float*)d_ws;
  float* W      = ws + OFF_W;
  float* M_all  = ws + OFF_M;
  float* S_all  = ws + OFF_S;
  float* out2   = ws + OFF_O2;
  float* y_part = ws + OFF_YP;
  float* out = (float*)d_out;

  hipLaunchKernelGGL(k_build_w, dim3(4096), dim3(256), 0, stream,
                     base_w, lora_A, lora_B, W);
  hipLaunchKernelGGL(k_gen_ms, dim3(NT, G), dim3(256), 0, stream,
                     query, cpg_w, cpg_b, ppg_w, ppg_b, ml, mr, sl, sr,
                     M_all, S_all);
  hipLaunchKernelGGL(k_mix, dim3(BN, G), dim3(256), 0, stream,
                     x, M_all, S_all, out2);
  hipLaunchKernelGGL(k_proj, dim3(NT, KC_N), dim3(256), 0, stream,
                     out2, W, y_part);
  hipLaunchKernelGGL(k_final, dim3(BN), dim3(256), 0, stream,
                     query, obias, gamma, beta, y_part, out);
}